// JacobiIterationLayer_43722767073700
// MI455X (gfx1250) — compile-verified
//
#include <hip/hip_runtime.h>
#include <cstdint>

// ---------------------------------------------------------------------------
// Jacobi sweep for the 4th-order cross stencil, LDS-tiled, using the CDNA5
// async global->LDS DMA path (ASYNCcnt-tracked) for tile staging.
//
// Memory-bound problem: ~16 flop / 12 B per point. 192 MB working set fits
// the MI455X 192 MB L2, so 5 separate sweeps stream from L2 after sweep 1.
// ---------------------------------------------------------------------------

#define TILE 32
#define HALO 2
#define LW   (TILE + 2 * HALO)   // 36 : staged tile edge
#define LP   (LW + 1)            // 37 : padded LDS pitch

// Async copy one f32 from global memory into LDS (gfx1250 VGLOBAL opcode 96,
// GLOBAL_LOAD_ASYNC_TO_LDS_B32, GV addressing). LDS destination address is a
// per-lane byte offset in a VGPR; completion is tracked with ASYNCcnt.
__device__ __forceinline__ void async_g2l_b32(uint32_t lds_byte_off,
                                              const float* gaddr) {
  asm volatile("global_load_async_to_lds_b32 %0, %1, off"
               :
               : "v"(lds_byte_off), "v"(gaddr)
               : "memory");
}

__device__ __forceinline__ void wait_async0() {
#if __has_builtin(__builtin_amdgcn_s_wait_asynccnt)
  __builtin_amdgcn_s_wait_asynccnt(0);
#else
  asm volatile("s_wait_asynccnt 0x0" ::: "memory");
#endif
}

__launch_bounds__(256)
__global__ void jacobi_step(const float* __restrict__ src,
                            const float* __restrict__ rhs,
                            const float* __restrict__ dx,
                            float* __restrict__ dst,
                            int H, int W) {
  __shared__ float tile[LW * LP];

  const int b  = blockIdx.z;
  const int y0 = blockIdx.y * TILE;
  const int x0 = blockIdx.x * TILE;
  const int tx = threadIdx.x;            // 0..31
  const int ty = threadIdx.y;            // 0..7
  const int tid = ty * 32 + tx;          // 0..255

  const float* srcb = src + (size_t)b * H * W;
  const float* rhsb = rhs + (size_t)b * H * W;
  float*       dstb = dst + (size_t)b * H * W;

  // AS3 pointer value (= byte offset from the workgroup LDS base) of tile[0].
  const uint32_t lds_base = (uint32_t)(uintptr_t)(&tile[0]);

  // ---- Stage the (TILE+4)^2 input tile into LDS via async DMA. -------------
  // Edge-clamped halo: clamped slots are only ever consumed by boundary
  // outputs, which are overwritten by the frozen-ring copy below.
  for (int i = tid; i < LW * LW; i += 256) {
    const int lr = i / LW;
    const int lc = i - lr * LW;
    int gy = y0 + lr - HALO; gy = gy < 0 ? 0 : (gy > H - 1 ? H - 1 : gy);
    int gx = x0 + lc - HALO; gx = gx < 0 ? 0 : (gx > W - 1 ? W - 1 : gx);
    async_g2l_b32(lds_base + (uint32_t)(lr * LP + lc) * 4u,
                  srcb + (size_t)gy * W + gx);
  }
  wait_async0();      // this wave's async writes to LDS are complete
  __syncthreads();    // all waves' tiles visible

  // ---- Per-batch scalars (uniform within the block -> scalar loads). -------
  const float dy = dx[b * 2 + 0];        // dim 0 = H
  const float dw = dx[b * 2 + 1];        // dim 1 = W
  const float sy = 1.0f / (dy * dy);
  const float sx = 1.0f / (dw * dw);
  const float dinv = 1.0f / (-2.5f * (sy + sx));
  const float w1 = 4.0f / 3.0f;
  const float w2 = -1.0f / 12.0f;

  // ---- 4 output rows per thread (32x32 tile / 256 threads). ----------------
#pragma unroll
  for (int k = 0; k < 4; ++k) {
    const int r  = ty + 8 * k;           // 0..31
    const int gy = y0 + r;
    const int gx = x0 + tx;
    const int lr = r + HALO;
    const int lc = tx + HALO;

    const float center = tile[lr * LP + lc];
    float out;
    if (gy < HALO || gy >= H - HALO || gx < HALO || gx >= W - HALO) {
      out = center;                      // frozen Dirichlet ring (g * mask)
    } else {
      const float cr =
          sy * (w2 * (tile[(lr - 2) * LP + lc] + tile[(lr + 2) * LP + lc]) +
                w1 * (tile[(lr - 1) * LP + lc] + tile[(lr + 1) * LP + lc])) +
          sx * (w2 * (tile[lr * LP + lc - 2] + tile[lr * LP + lc + 2]) +
                w1 * (tile[lr * LP + lc - 1] + tile[lr * LP + lc + 1]));
      out = dinv * (rhsb[(size_t)gy * W + gx] - cr);
    }
    dstb[(size_t)gy * W + gx] = out;
  }
}

extern "C" void kernel_launch(void* const* d_in, const int* in_sizes, int n_in,
                              void* d_out, int out_size, void* d_ws, size_t ws_size,
                              hipStream_t stream) {
  const float* guess = (const float*)d_in[0];  // (16,1,1024,1024) f32
  const float* rhs   = (const float*)d_in[1];  // (16,1,1024,1024) f32
  const float* dxv   = (const float*)d_in[2];  // (16,2) f32

  float* out = (float*)d_out;
  float* tmp = (float*)d_ws;                   // needs 16*1024*1024*4 = 64 MiB

  const int B = in_sizes[2] / 2;               // 16
  const int H = 1024;
  const int W = (in_sizes[0] / B) / H;         // 1024

  dim3 block(32, 8);                           // 8 wave32s per block
  dim3 grid(W / TILE, H / TILE, B);

  // 5 Jacobi sweeps; ping-pong so the final sweep lands in d_out.
  jacobi_step<<<grid, block, 0, stream>>>(guess, rhs, dxv, out, H, W);
  jacobi_step<<<grid, block, 0, stream>>>(out,   rhs, dxv, tmp, H, W);
  jacobi_step<<<grid, block, 0, stream>>>(tmp,   rhs, dxv, out, H, W);
  jacobi_step<<<grid, block, 0, stream>>>(out,   rhs, dxv, tmp, H, W);
  jacobi_step<<<grid, block, 0, stream>>>(tmp,   rhs, dxv, out, H, W);
}